// NetGCN_23682449670433
// MI455X (gfx1250) — compile-verified
//
#include <hip/hip_runtime.h>
#include <hip/hip_bf16.h>

#define N_NODES 100000
#define N_EDGES 1600000
#define DIM     128
#define BN_EPS  1e-5f

typedef float v2f __attribute__((ext_vector_type(2)));
typedef float v8f __attribute__((ext_vector_type(8)));

// Hardware float atomic-add (GLOBAL_ATOMIC_ADD_F32, no-return -> STOREcnt).
// agg lives in coarse-grained hipMalloc'd memory, so the "unsafe" (non-
// fine-grained) hardware path is valid.
__device__ __forceinline__ void fatomic_add(float* addr, float v) {
#if defined(__HIP_DEVICE_COMPILE__)
    unsafeAtomicAdd(addr, v);
#else
    atomicAdd(addr, v);
#endif
}

// ---------------------------------------------------------------------------
// Utility: zero a float buffer
// ---------------------------------------------------------------------------
__global__ void k_zero(float* __restrict__ p, long long n) {
    long long i = (long long)blockIdx.x * blockDim.x + threadIdx.x;
    if (i < n) p[i] = 0.0f;
}

// ---------------------------------------------------------------------------
// Degree count: deg[row[e]] += 1   (deg buffer pre-zeroed)
// ---------------------------------------------------------------------------
__global__ void k_deg(const int* __restrict__ row, float* __restrict__ deg, int E) {
    int i = blockIdx.x * blockDim.x + threadIdx.x;
    if (i < E) fatomic_add(&deg[row[i]], 1.0f);
}

// dinv[i] = rsqrt(deg[i] + 1)   (in place)
__global__ void k_dinv(float* __restrict__ deg, int n) {
    int i = blockIdx.x * blockDim.x + threadIdx.x;
    if (i < n) deg[i] = rsqrtf(deg[i] + 1.0f);
}

// norm[e] = dinv[row[e]] * dinv[col[e]]
__global__ void k_norm(const int* __restrict__ row, const int* __restrict__ col,
                       const float* __restrict__ dinv, float* __restrict__ nrm, int E) {
    int i = blockIdx.x * blockDim.x + threadIdx.x;
    if (i < E) nrm[i] = dinv[row[i]] * dinv[col[i]];
}

// ---------------------------------------------------------------------------
// Dense GEMM: out[M,128] = A[M,128] @ W[128,128] + bias[128]
// fp32 WMMA 16x16x4.  One wave -> one 16x16 C tile; block = 8 waves = 16x128.
// A fragment (16x4 f32, 2 VGPR): lanes 0-15 hold K=k,k+1 of row M=lane;
//                                lanes 16-31 hold K=k+2,k+3 of row M=lane-16.
// B fragment (4x16 f32, 2 VGPR): lanes 0-15 hold rows K=k,k+1 at N=lane;
//                                lanes 16-31 hold rows K=k+2,k+3 at N=lane-16.
// C/D (16x16 f32, 8 VGPR): vgpr i, lanes 0-15 -> M=i, lanes 16-31 -> M=8+i, N=lane%16.
// ---------------------------------------------------------------------------
__global__ void k_gemm(const float* __restrict__ A, const float* __restrict__ W,
                       const float* __restrict__ bias, float* __restrict__ out) {
    const int wave = threadIdx.x >> 5;      // 0..7 -> column tile
    const int lane = threadIdx.x & 31;
    const int hi   = lane >> 4;             // 0: lanes 0-15, 1: lanes 16-31
    const int l    = lane & 15;
    const int row0 = blockIdx.x * 16;
    const int col0 = wave * 16;

    const float* arow = A + (long long)(row0 + l) * DIM;

    v8f acc = {};
#pragma unroll
    for (int k = 0; k < DIM; k += 4) {
        const int ka = k + hi * 2;
        v2f a = *(const v2f*)(arow + ka);                 // A[M=row0+l][ka, ka+1]
        v2f b;
        b.x = W[(long long)ka * DIM + col0 + l];          // B[K=ka  ][N=col0+l]
        b.y = W[(long long)(ka + 1) * DIM + col0 + l];    // B[K=ka+1][N=col0+l]
        acc = __builtin_amdgcn_wmma_f32_16x16x4_f32(
            /*neg_a=*/false, a, /*neg_b=*/false, b,
            /*c_mod=*/(short)0, acc, /*reuse_a=*/false, /*reuse_b=*/false);
    }

    const int n = col0 + l;
    const float bn = bias[n];
#pragma unroll
    for (int i = 0; i < 8; ++i) {
        const int m = row0 + hi * 8 + i;
        out[(long long)m * DIM + n] = acc[i] + bn;
    }
}

// ---------------------------------------------------------------------------
// Fused edge message: for edge e,
//   msg = norm[e] * relu( h[row[e]] + edge_attr[e] @ We + be )
//   agg[col[e]] += msg
// One wave per edge, 4 channels per lane (float4).  Edge-linear (K=3) fused.
// ---------------------------------------------------------------------------
__global__ void k_edge(const float* __restrict__ h, const int* __restrict__ row,
                       const int* __restrict__ col, const float* __restrict__ ea,
                       const float* __restrict__ We, const float* __restrict__ be,
                       const float* __restrict__ nrm, float* __restrict__ agg) {
    const int lane = threadIdx.x & 31;
    const int e = blockIdx.x * (blockDim.x >> 5) + (threadIdx.x >> 5);
    if (e >= N_EDGES) return;

    const int r = row[e];
    const int c = col[e];
    const float nm = nrm[e];
    const float a0 = ea[3 * e + 0];
    const float a1 = ea[3 * e + 1];
    const float a2 = ea[3 * e + 2];

    const int j = lane * 4;
    const float4 hv = *(const float4*)(h + (long long)r * DIM + j);
    const float4 w0 = *(const float4*)(We + 0 * DIM + j);
    const float4 w1 = *(const float4*)(We + 1 * DIM + j);
    const float4 w2 = *(const float4*)(We + 2 * DIM + j);
    const float4 bb = *(const float4*)(be + j);

    float4 m;
    m.x = nm * fmaxf(hv.x + bb.x + a0 * w0.x + a1 * w1.x + a2 * w2.x, 0.0f);
    m.y = nm * fmaxf(hv.y + bb.y + a0 * w0.y + a1 * w1.y + a2 * w2.y, 0.0f);
    m.z = nm * fmaxf(hv.z + bb.z + a0 * w0.z + a1 * w1.z + a2 * w2.z, 0.0f);
    m.w = nm * fmaxf(hv.w + bb.w + a0 * w0.w + a1 * w1.w + a2 * w2.w, 0.0f);

    float* dst = agg + (long long)c * DIM + j;
    fatomic_add(dst + 0, m.x);
    fatomic_add(dst + 1, m.y);
    fatomic_add(dst + 2, m.z);
    fatomic_add(dst + 3, m.w);
}

// ---------------------------------------------------------------------------
// BN stats on relu(agg): sums[c] = sum, sums[128+c] = sumsq  (pre-zeroed)
// 128 threads/block, thread = column (coalesced), 256 rows per block.
// ---------------------------------------------------------------------------
__global__ void k_bn_stats(const float* __restrict__ agg, float* __restrict__ sums) {
    const int c = threadIdx.x;              // 0..127
    const int r0 = blockIdx.x * 256;
    const int r1 = (r0 + 256 < N_NODES) ? r0 + 256 : N_NODES;
    float s = 0.0f, sq = 0.0f;
    for (int r = r0; r < r1; ++r) {
        float v = fmaxf(agg[(long long)r * DIM + c], 0.0f);
        s += v;
        sq += v * v;
    }
    fatomic_add(&sums[c], s);
    fatomic_add(&sums[DIM + c], sq);
}

// out[idx] = (relu(agg[idx]) - mean) * rsqrt(var + eps) * g + bt  (in place ok)
__global__ void k_bn_apply(const float* __restrict__ agg, const float* __restrict__ sums,
                           const float* __restrict__ g, const float* __restrict__ bt,
                           float* __restrict__ out) {
    const long long idx = (long long)blockIdx.x * blockDim.x + threadIdx.x;
    if (idx >= (long long)N_NODES * DIM) return;
    const int c = (int)(idx & (DIM - 1));
    const float inv_n = 1.0f / (float)N_NODES;
    const float mean = sums[c] * inv_n;
    const float var = sums[DIM + c] * inv_n - mean * mean;
    const float v = fmaxf(agg[idx], 0.0f);
    out[idx] = (v - mean) * rsqrtf(var + BN_EPS) * g[c] + bt[c];
}

// ---------------------------------------------------------------------------
// Host-side launcher
// ---------------------------------------------------------------------------
extern "C" void kernel_launch(void* const* d_in, const int* in_sizes, int n_in,
                              void* d_out, int out_size, void* d_ws, size_t ws_size,
                              hipStream_t stream) {
    (void)in_sizes; (void)n_in; (void)out_size; (void)ws_size;

    const float* x  = (const float*)d_in[0];
    const int*   ei = (const int*)d_in[1];
    const float* ea = (const float*)d_in[2];
    const int* row = ei;
    const int* col = ei + N_EDGES;

    const float* W[3]  = { (const float*)d_in[3],  (const float*)d_in[9],  (const float*)d_in[15] };
    const float* b[3]  = { (const float*)d_in[4],  (const float*)d_in[10], (const float*)d_in[16] };
    const float* We[3] = { (const float*)d_in[5],  (const float*)d_in[11], (const float*)d_in[17] };
    const float* be[3] = { (const float*)d_in[6],  (const float*)d_in[12], (const float*)d_in[18] };
    const float* g[3]  = { (const float*)d_in[7],  (const float*)d_in[13], (const float*)d_in[19] };
    const float* bt[3] = { (const float*)d_in[8],  (const float*)d_in[14], (const float*)d_in[20] };
    const float* Wl = (const float*)d_in[21];
    const float* bl = (const float*)d_in[22];

    // Workspace layout (floats)
    const long long ND = (long long)N_NODES * DIM;   // 12.8M
    float* ws   = (float*)d_ws;
    float* hlin = ws;                 // [N, 128]
    float* agg  = ws + ND;            // [N, 128]  (BN applied in place -> next act)
    float* dinv = ws + 2 * ND;        // [N]
    float* nrm  = dinv + N_NODES;     // [E]
    float* sums = nrm + N_EDGES;      // [256]

    const int T = 256;
    const int gN    = (N_NODES + T - 1) / T;                 // 391
    const int gE    = (N_EDGES + T - 1) / T;                 // 6250
    const int gND   = (int)((ND + T - 1) / T);               // 50000
    const int gEdge = N_EDGES / (T / 32);                    // 200000 (exact)
    const int gTile = N_NODES / 16;                          // 6250   (exact)
    const int gStat = (N_NODES + 255) / 256;                 // 391

    // --- degree / symmetric normalization (depends only on edge_index) ---
    k_zero<<<gN, T, 0, stream>>>(dinv, N_NODES);
    k_deg <<<gE, T, 0, stream>>>(row, dinv, N_EDGES);
    k_dinv<<<gN, T, 0, stream>>>(dinv, N_NODES);
    k_norm<<<gE, T, 0, stream>>>(row, col, dinv, nrm, N_EDGES);

    // --- 3 GCN layers ---
    const float* act = x;
    for (int L = 0; L < 3; ++L) {
        k_gemm<<<gTile, T, 0, stream>>>(act, W[L], b[L], hlin);
        k_zero<<<gND, T, 0, stream>>>(agg, ND);
        k_edge<<<gEdge, T, 0, stream>>>(hlin, row, col, ea, We[L], be[L], nrm, agg);
        k_zero<<<1, T, 0, stream>>>(sums, 2 * DIM);
        k_bn_stats<<<gStat, DIM, 0, stream>>>(agg, sums);
        k_bn_apply<<<gND, T, 0, stream>>>(agg, sums, g[L], bt[L], agg);
        act = agg;
    }

    // --- final projection ---
    k_gemm<<<gTile, T, 0, stream>>>(act, Wl, bl, (float*)d_out);
}